// DGM_Layer_5274219839836
// MI455X (gfx1250) — compile-verified
//
#include <hip/hip_runtime.h>
#include <math.h>

typedef float v2f __attribute__((ext_vector_type(2)));
typedef float v8f __attribute__((ext_vector_type(8)));

#define BATCH 8
#define NNODE 2048
#define DDIM  512
#define MROWS (BATCH * NNODE)   // 16384

// workspace layout (floats)
#define WS_CENTROID 0      // 512 floats (sum -> mean, finalized in place)
#define WS_MAXBITS  512    // 1 uint (max |x - c| as float bits)
#define WS_SQ       1024   // 16384 floats (per-row ||h||^2; scale^2 applied in epilogue)

static __device__ __forceinline__ v8f wmma_f32(v2f a, v2f b, v8f c) {
    // D = A(16x4) * B(4x16) + C  -- fp32 WMMA, wave32
    return __builtin_amdgcn_wmma_f32_16x16x4_f32(false, a, false, b, (short)0, c, false, false);
}

__global__ void k_init(float* ws) {
    int i = blockIdx.x * blockDim.x + threadIdx.x;
    if (i < 1024) ws[i] = 0.0f;   // zeros centroid sums + maxabs bits
}

// column sums of x [16384,512]: each block accumulates 64 rows in registers
__global__ void k_colsum(const float* __restrict__ x, float* __restrict__ csum) {
    const int t = threadIdx.x;            // 256 threads
    const int row0 = blockIdx.x * 64;
    float a0 = 0.f, a1 = 0.f;
    for (int r = 0; r < 64; ++r) {
        const float* xr = x + (size_t)(row0 + r) * DDIM;
        a0 += xr[t];
        a1 += xr[t + 256];
    }
    atomicAdd(&csum[t], a0);
    atomicAdd(&csum[t + 256], a1);
}

__global__ void k_finalize_centroid(float* c) {
    c[threadIdx.x] *= (1.0f / (float)MROWS);   // 512 threads
}

// global max of |x - centroid|; bit-pattern atomicMax is order-correct for >=0 floats
__global__ void k_maxabs(const float* __restrict__ x, const float* __restrict__ c,
                         unsigned* __restrict__ outbits) {
    __shared__ float s[256];
    const int t = threadIdx.x;
    size_t idx = (size_t)blockIdx.x * blockDim.x + t;
    const size_t stride = (size_t)gridDim.x * blockDim.x;
    float m = 0.f;
    for (size_t i = idx; i < (size_t)MROWS * DDIM; i += stride) {
        int col = (int)(i & (DDIM - 1));
        m = fmaxf(m, fabsf(x[i] - c[col]));
    }
    s[t] = m;
    __syncthreads();
    for (int o = 128; o > 0; o >>= 1) {
        if (t < o) s[t] = fmaxf(s[t], s[t + o]);
        __syncthreads();
    }
    if (t == 0) atomicMax(outbits, __float_as_uint(s[0]));
}

// GEMM1: h = relu(x @ W + b), M=16384 N=512 K=512
// 8 waves/block; wave tile 64x32 (4x2 WMMA tiles); block tile 128x128
__global__ __launch_bounds__(256) void k_gemm_h(const float* __restrict__ x,
                                                const float* __restrict__ W,
                                                const float* __restrict__ bias,
                                                float* __restrict__ h) {
    const int lane  = threadIdx.x & 31;
    const int wave  = threadIdx.x >> 5;
    const int waveM = wave & 1;
    const int waveN = wave >> 1;
    const int lm = lane & 15;
    const int lk = (lane >> 4) << 1;     // 0 or 2 (A/B K sub-index per ISA layout)
    const int rowBase = blockIdx.x * 128 + waveM * 64;
    const int colBase = blockIdx.y * 128 + waveN * 32;

    v8f zero;
#pragma unroll
    for (int r = 0; r < 8; ++r) zero[r] = 0.f;
    v8f acc[4][2];
#pragma unroll
    for (int mt = 0; mt < 4; ++mt)
#pragma unroll
        for (int nt = 0; nt < 2; ++nt) acc[mt][nt] = zero;

    for (int k0 = 0; k0 < DDIM; k0 += 4) {
        v2f a[4];
#pragma unroll
        for (int mt = 0; mt < 4; ++mt) {
            const float* p = x + (size_t)(rowBase + mt * 16 + lm) * DDIM + (k0 + lk);
            a[mt] = *(const v2f*)p;
        }
        v2f bf[2];
#pragma unroll
        for (int nt = 0; nt < 2; ++nt) {
            const int col = colBase + nt * 16 + lm;
            bf[nt].x = W[(size_t)(k0 + lk) * DDIM + col];
            bf[nt].y = W[(size_t)(k0 + lk + 1) * DDIM + col];
        }
#pragma unroll
        for (int mt = 0; mt < 4; ++mt)
#pragma unroll
            for (int nt = 0; nt < 2; ++nt)
                acc[mt][nt] = wmma_f32(a[mt], bf[nt], acc[mt][nt]);
    }

    const int hi8 = (lane >> 4) << 3;
#pragma unroll
    for (int nt = 0; nt < 2; ++nt) {
        const int col = colBase + nt * 16 + lm;
        const float bv = bias[col];
#pragma unroll
        for (int mt = 0; mt < 4; ++mt) {
#pragma unroll
            for (int r = 0; r < 8; ++r) {
                const int row = rowBase + mt * 16 + hi8 + r;
                float v = acc[mt][nt][r] + bv;
                h[(size_t)row * DDIM + col] = v > 0.f ? v : 0.f;
            }
        }
    }
}

// per-row ||h||^2 (centroid cancels in pairwise distances; scale^2 applied later)
__global__ void k_rowsq(const float* __restrict__ h, float* __restrict__ hsq) {
    __shared__ float s[256];
    const int row = blockIdx.x;
    const int t = threadIdx.x;
    v2f hv = *(const v2f*)&h[(size_t)row * DDIM + 2 * t];
    s[t] = hv.x * hv.x + hv.y * hv.y;
    __syncthreads();
    for (int o = 128; o > 0; o >>= 1) {
        if (t < o) s[t] += s[t + o];
        __syncthreads();
    }
    if (t == 0) hsq[row] = s[0];
}

// GEMM2 per batch: G = h_b @ h_b^T (pure load+wmma inner loop);
// epilogue: Dmat = scale^2*(hsq_i + hsq_j - 2G), A = sigmoid(t*(thr - Dmat)), mask = A>0.5
__global__ __launch_bounds__(256) void k_gram(const float* __restrict__ h,
                                              const float* __restrict__ ws,
                                              const float* __restrict__ temp_p,
                                              const float* __restrict__ thr_p,
                                              float* __restrict__ Aout,
                                              float* __restrict__ Mout) {
    const float* hsq = ws + WS_SQ;
    const unsigned* mb = (const unsigned*)(ws + WS_MAXBITS);
    const float scale = 0.9f / __uint_as_float(*mb);
    const float scale2 = scale * scale;

    const int batch = blockIdx.z;
    const int lane  = threadIdx.x & 31;
    const int wave  = threadIdx.x >> 5;
    const int waveM = wave & 1;
    const int waveN = wave >> 1;
    const int lm = lane & 15;
    const int lk = (lane >> 4) << 1;
    const int rowBase = blockIdx.x * 128 + waveM * 64;   // within batch
    const int colBase = blockIdx.y * 128 + waveN * 32;
    const float* hb = h + (size_t)batch * NNODE * DDIM;

    v8f zero;
#pragma unroll
    for (int r = 0; r < 8; ++r) zero[r] = 0.f;
    v8f acc[4][2];
#pragma unroll
    for (int mt = 0; mt < 4; ++mt)
#pragma unroll
        for (int nt = 0; nt < 2; ++nt) acc[mt][nt] = zero;

    for (int k0 = 0; k0 < DDIM; k0 += 4) {
        if (k0 + 64 < DDIM)   // prefetch A-row data ahead
            __builtin_prefetch(&hb[(size_t)(rowBase + lm) * DDIM + k0 + lk + 64], 0, 0);
        v2f a[4];
#pragma unroll
        for (int mt = 0; mt < 4; ++mt)
            a[mt] = *(const v2f*)&hb[(size_t)(rowBase + mt * 16 + lm) * DDIM + k0 + lk];
        v2f bf[2];
#pragma unroll
        for (int nt = 0; nt < 2; ++nt)
            bf[nt] = *(const v2f*)&hb[(size_t)(colBase + nt * 16 + lm) * DDIM + k0 + lk];
#pragma unroll
        for (int mt = 0; mt < 4; ++mt)
#pragma unroll
            for (int nt = 0; nt < 2; ++nt)
                acc[mt][nt] = wmma_f32(a[mt], bf[nt], acc[mt][nt]);
    }

    const float temp = *temp_p;
    const float thr  = *thr_p;
    const int hi8 = (lane >> 4) << 3;
#pragma unroll
    for (int nt = 0; nt < 2; ++nt) {
        const int j = colBase + nt * 16 + lm;
        const float sqj = hsq[batch * NNODE + j];
#pragma unroll
        for (int mt = 0; mt < 4; ++mt) {
#pragma unroll
            for (int r = 0; r < 8; ++r) {
                const int i = rowBase + mt * 16 + hi8 + r;
                const float sqi = hsq[batch * NNODE + i];
                const float dm = scale2 * (sqi + sqj - 2.0f * acc[mt][nt][r]);
                const float z = temp * (thr - dm);
                float av = 1.0f / (1.0f + __expf(-z));
                if (i == j) av = 0.f;   // diag: -inf then clamp(min=0) -> exactly 0
                const size_t off = ((size_t)batch * NNODE + i) * NNODE + j;
                Aout[off] = av;
                Mout[off] = (av > 0.5f) ? 1.0f : 0.0f;
            }
        }
    }
}

extern "C" void kernel_launch(void* const* d_in, const int* in_sizes, int n_in,
                              void* d_out, int out_size, void* d_ws, size_t ws_size,
                              hipStream_t stream) {
    (void)in_sizes; (void)n_in; (void)out_size; (void)ws_size;
    const float* x    = (const float*)d_in[0];
    const float* W    = (const float*)d_in[1];
    const float* bias = (const float*)d_in[2];
    const float* temp = (const float*)d_in[3];
    const float* thr  = (const float*)d_in[4];

    float* out  = (float*)d_out;
    float* h    = out;                                   // x_out [8,2048,512]
    float* Aout = out + (size_t)MROWS * DDIM;            // A     [8,2048,2048]
    float* Mout = Aout + (size_t)BATCH * NNODE * NNODE;  // mask  [8,2048,2048]
    float* ws   = (float*)d_ws;

    k_init<<<4, 256, 0, stream>>>(ws);
    k_colsum<<<MROWS / 64, 256, 0, stream>>>(x, ws + WS_CENTROID);
    k_finalize_centroid<<<1, 512, 0, stream>>>(ws + WS_CENTROID);
    k_maxabs<<<512, 256, 0, stream>>>(x, ws + WS_CENTROID, (unsigned*)(ws + WS_MAXBITS));
    k_gemm_h<<<dim3(MROWS / 128, DDIM / 128), 256, 0, stream>>>(x, W, bias, h);
    k_rowsq<<<MROWS, 256, 0, stream>>>(h, ws + WS_SQ);
    k_gram<<<dim3(NNODE / 128, NNODE / 128, BATCH), 256, 0, stream>>>(h, ws, temp, thr, Aout, Mout);
}